// HarmonicLatticePotential_6313601925565
// MI455X (gfx1250) — compile-verified
//
#include <hip/hip_runtime.h>

#define K_CONST   1.0f
#define R0_CONST  1.0f
#define EPS_CONST 1e-20f
#define NUM_GRAPHS 64

// Native clang vector => legal operand for __builtin_nontemporal_load,
// lowers to one global_load_b128 with NT temporal hint.
typedef int vint4 __attribute__((ext_vector_type(4)));

// 12-byte position record; 4-byte aligned, clang emits b96/b64+b32 vector loads.
struct Float3 { float x, y, z; };

__global__ void hlp_zero_kernel(float* __restrict__ out, int n) {
  int stride = gridDim.x * blockDim.x;
  for (int i = blockIdx.x * blockDim.x + threadIdx.x; i < n; i += stride)
    out[i] = 0.0f;
}

// One edge: gather, compute, scatter. Fire-and-forget f32 atomics
// (result unused => non-returning global_atomic_add_f32, STOREcnt-tracked).
__device__ __forceinline__ void hlp_process_edge(const float* __restrict__ pos,
                                                 const int*   __restrict__ batch,
                                                 float*       __restrict__ forces,
                                                 float*       __restrict__ bins,
                                                 int i, int j) {
  const Float3 pi = *(const Float3*)(pos + 3 * i);   // L2-resident gathers
  const Float3 pj = *(const Float3*)(pos + 3 * j);

  const float dx = pi.x - pj.x;
  const float dy = pi.y - pj.y;
  const float dz = pi.z - pj.z;
  const float d2 = dx * dx + dy * dy + dz * dz;
  const float d  = __builtin_sqrtf(d2);              // v_sqrt_f32
  const float delta = d - R0_CONST;

  // energy contribution -> per-block LDS bin (ds_add_f32, no return)
  const float ee = 0.5f * K_CONST * delta * delta;
  atomicAdd(&bins[batch[i]], ee);

  const float s  = (K_CONST * delta) * __builtin_amdgcn_rcpf(d + EPS_CONST); // v_rcp_f32
  const float fx = s * dx, fy = s * dy, fz = s * dz;

  float* fi = forces + 3 * i;
  float* fj = forces + 3 * j;
  __hip_atomic_fetch_add(fi + 0, -fx, __ATOMIC_RELAXED, __HIP_MEMORY_SCOPE_AGENT);
  __hip_atomic_fetch_add(fi + 1, -fy, __ATOMIC_RELAXED, __HIP_MEMORY_SCOPE_AGENT);
  __hip_atomic_fetch_add(fi + 2, -fz, __ATOMIC_RELAXED, __HIP_MEMORY_SCOPE_AGENT);
  __hip_atomic_fetch_add(fj + 0,  fx, __ATOMIC_RELAXED, __HIP_MEMORY_SCOPE_AGENT);
  __hip_atomic_fetch_add(fj + 1,  fy, __ATOMIC_RELAXED, __HIP_MEMORY_SCOPE_AGENT);
  __hip_atomic_fetch_add(fj + 2,  fz, __ATOMIC_RELAXED, __HIP_MEMORY_SCOPE_AGENT);
}

// Vectorized index stream: one NT global_load_b128 per row per 4 edges;
// 4 independent edge chains per iteration for memory-level parallelism.
__global__ __launch_bounds__(256)
void hlp_edge_kernel_v4(const float* __restrict__ pos,
                        const int*   __restrict__ ei,
                        const int*   __restrict__ ej,
                        const int*   __restrict__ batch,
                        float*       __restrict__ energy,
                        float*       __restrict__ forces,
                        int E) {
  __shared__ float bins[NUM_GRAPHS];
  if (threadIdx.x < NUM_GRAPHS) bins[threadIdx.x] = 0.0f;
  __syncthreads();

  const int tid      = blockIdx.x * blockDim.x + threadIdx.x;
  const int nthreads = gridDim.x * blockDim.x;
  const int E4       = E >> 2;

  const vint4* ei4 = (const vint4*)ei;
  const vint4* ej4 = (const vint4*)ej;

  for (int g = tid; g < E4; g += nthreads) {
    // Prefetch next stride of both streamed rows (global_prefetch_b8;
    // speculative => past-the-end translation failures silently dropped).
    __builtin_prefetch(ei4 + g + nthreads, 0, 1);
    __builtin_prefetch(ej4 + g + nthreads, 0, 1);

    // Streamed once: NT hint keeps near caches warm for pos/forces.
    const vint4 vi = __builtin_nontemporal_load(ei4 + g);
    const vint4 vj = __builtin_nontemporal_load(ej4 + g);

    hlp_process_edge(pos, batch, forces, bins, vi.x, vj.x);
    hlp_process_edge(pos, batch, forces, bins, vi.y, vj.y);
    hlp_process_edge(pos, batch, forces, bins, vi.z, vj.z);
    hlp_process_edge(pos, batch, forces, bins, vi.w, vj.w);
  }

  // Tail (empty when E % 4 == 0, kept for generality).
  for (int e = (E4 << 2) + tid; e < E; e += nthreads)
    hlp_process_edge(pos, batch, forces, bins, ei[e], ej[e]);

  __syncthreads();
  if (threadIdx.x < NUM_GRAPHS) {
    const float v = bins[threadIdx.x];
    if (v != 0.0f)
      __hip_atomic_fetch_add(&energy[threadIdx.x], v,
                             __ATOMIC_RELAXED, __HIP_MEMORY_SCOPE_AGENT);
  }
}

// Scalar fallback (used only if ej row would be misaligned for int4 groups,
// i.e. E % 4 != 0).
__global__ __launch_bounds__(256)
void hlp_edge_kernel_s(const float* __restrict__ pos,
                       const int*   __restrict__ ei,
                       const int*   __restrict__ ej,
                       const int*   __restrict__ batch,
                       float*       __restrict__ energy,
                       float*       __restrict__ forces,
                       int E) {
  __shared__ float bins[NUM_GRAPHS];
  if (threadIdx.x < NUM_GRAPHS) bins[threadIdx.x] = 0.0f;
  __syncthreads();

  const int stride = gridDim.x * blockDim.x;
  for (int e = blockIdx.x * blockDim.x + threadIdx.x; e < E; e += stride) {
    const int i = __builtin_nontemporal_load(ei + e);
    const int j = __builtin_nontemporal_load(ej + e);
    hlp_process_edge(pos, batch, forces, bins, i, j);
  }

  __syncthreads();
  if (threadIdx.x < NUM_GRAPHS) {
    const float v = bins[threadIdx.x];
    if (v != 0.0f)
      __hip_atomic_fetch_add(&energy[threadIdx.x], v,
                             __ATOMIC_RELAXED, __HIP_MEMORY_SCOPE_AGENT);
  }
}

extern "C" void kernel_launch(void* const* d_in, const int* in_sizes, int n_in,
                              void* d_out, int out_size, void* d_ws, size_t ws_size,
                              hipStream_t stream) {
  (void)n_in; (void)d_ws; (void)ws_size;

  const float* pos   = (const float*)d_in[0];           // (N,3) f32
  const int*   eidx  = (const int*)  d_in[1];           // (2,E) int
  const int*   batch = (const int*)  d_in[2];           // (N,)  int

  const int E = in_sizes[1] / 2;

  float* out    = (float*)d_out;
  float* energy = out;                 // first 64 floats
  float* forces = out + NUM_GRAPHS;    // then N*3 floats

  // Zero outputs every launch (harness poisons d_out, never re-poisons).
  {
    int n = out_size;
    int blocks = (n + 255) / 256;
    hlp_zero_kernel<<<blocks, 256, 0, stream>>>(out, n);
  }

  // Fixed modest grid: 1536 blocks * 8 wave32 waves saturates the WGPs while
  // keeping the per-bin energy flush at 1536 same-address L2 atomics total.
  const int blocks = 1536;
  if ((E & 3) == 0) {
    hlp_edge_kernel_v4<<<blocks, 256, 0, stream>>>(pos, eidx, eidx + E, batch,
                                                   energy, forces, E);
  } else {
    hlp_edge_kernel_s<<<blocks, 256, 0, stream>>>(pos, eidx, eidx + E, batch,
                                                  energy, forces, E);
  }
}